// RGCN_LP_27917287424410
// MI455X (gfx1250) — compile-verified
//
#include <hip/hip_runtime.h>
#include <hip/hip_bf16.h>

// RGCN link prediction on MI455X (gfx1250, wave32).
// All dense math runs on V_WMMA_F32_16X16X4_F32 (full-fp32 tensor op).
// Mean-aggregation is linear, so we scatter-add RAW features per (rel,dst),
// scale by 1/max(cnt,1) (folded into the WMMA A-fragment load) and do one
// dense GEMM per relation accumulated into the output via the WMMA C operand.

typedef float v2f __attribute__((ext_vector_type(2)));
typedef float v8f __attribute__((ext_vector_type(8)));

#define N_NODES 100000
#define N_REL   8
#define N_BASES 30
#define N_EDGES 640000
#define N_PAIRS 200000

// ---------------------------------------------------------------------------
// Generic fp32 WMMA GEMM:  C[M,N] (=|+=) rowscale(A[M,K]) @ B[K,N] (+bias)(relu)
// One wave computes a 16-row x N strip. NT = N/16 accumulator tiles.
// blockDim = 128 (4 waves). row0 is wave-uniform, so the early-out keeps
// EXEC all-ones for surviving waves (WMMA requirement).
// A-fragment (16x4 f32): lane m = M row, VGPR j holds K = j + 2*(lane>>4).
// C/D (16x16 f32): VGPR v, lanes 0-15 -> M=v, lanes 16-31 -> M=v+8, N=lane&15.
// ---------------------------------------------------------------------------
template <int NT>
__global__ void __launch_bounds__(128)
gemm_wmma_f32(const float* __restrict__ A, const float* __restrict__ B,
              const float* __restrict__ bias, const float* __restrict__ rowscale,
              float* __restrict__ C, int M, int K, int accumulate, int relu)
{
    const int N    = NT * 16;
    const int wave = threadIdx.x >> 5;
    const int lane = threadIdx.x & 31;
    const int row0 = (blockIdx.x * 4 + wave) * 16;
    if (row0 >= M) return;                 // wave-uniform exit

    const int m  = lane & 15;              // M within tile (A frag / C column map)
    const int kh = lane >> 4;              // K-half select (0/1)
    const int n  = lane & 15;              // N within tile (B frag / C)
    const int mrow = row0 + m;

    v8f acc[NT];
#pragma unroll
    for (int t = 0; t < NT; ++t) {
        v8f c;
        if (accumulate) {
#pragma unroll
            for (int v = 0; v < 8; ++v)
                c[v] = C[(size_t)(row0 + v + kh * 8) * N + t * 16 + n];
        } else if (bias) {
            const float bv = bias[t * 16 + n];
#pragma unroll
            for (int v = 0; v < 8; ++v) c[v] = bv;
        } else {
#pragma unroll
            for (int v = 0; v < 8; ++v) c[v] = 0.f;
        }
        acc[t] = c;
    }

    const float  rs   = rowscale ? rowscale[mrow] : 1.0f;   // mean division folded in
    const float* Arow = A + (size_t)mrow * K;

    for (int k = 0; k < K; k += 4) {
        v2f a;
        a.x = Arow[k + kh * 2 + 0] * rs;
        a.y = Arow[k + kh * 2 + 1] * rs;
#pragma unroll
        for (int t = 0; t < NT; ++t) {
            const float* Bp = B + (size_t)(k + kh * 2) * N + t * 16 + n;
            v2f b;
            b.x = Bp[0];
            b.y = Bp[N];
            acc[t] = __builtin_amdgcn_wmma_f32_16x16x4_f32(
                false, a, false, b, (short)0, acc[t], false, false);
        }
    }

#pragma unroll
    for (int t = 0; t < NT; ++t) {
#pragma unroll
        for (int v = 0; v < 8; ++v) {
            float val = acc[t][v];
            if (relu) val = fmaxf(val, 0.f);
            C[(size_t)(row0 + v + kh * 8) * N + t * 16 + n] = val;
        }
    }
}

// W[r] = sum_b comp[r,b] * basis[b]   (tiny einsum, R*IO elements)
__global__ void compute_rel_weights(const float* __restrict__ comp,
                                    const float* __restrict__ basis,
                                    float* __restrict__ W, int R, int NB, int IO)
{
    int idx = blockIdx.x * blockDim.x + threadIdx.x;
    if (idx >= R * IO) return;
    int r = idx / IO, j = idx - r * IO;
    float s = 0.f;
    for (int b = 0; b < NB; ++b)
        s += comp[r * NB + b] * basis[(size_t)b * IO + j];
    W[idx] = s;
}

__global__ void fill_zero(float* __restrict__ p, size_t n)
{
    size_t i = (size_t)blockIdx.x * blockDim.x + threadIdx.x;
    size_t stride = (size_t)gridDim.x * blockDim.x;
    for (; i < n; i += stride) p[i] = 0.f;
}

// per-(relation,dst) edge counts (shared by both convs)
__global__ void count_kernel(const int* __restrict__ dst, const int* __restrict__ et,
                             float* __restrict__ cnt, int E)
{
    int e = blockIdx.x * blockDim.x + threadIdx.x;
    if (e < E) atomicAdd(&cnt[(size_t)et[e] * N_NODES + dst[e]], 1.0f);
}

__global__ void invert_kernel(float* __restrict__ c, size_t n)
{
    size_t i = (size_t)blockIdx.x * blockDim.x + threadIdx.x;
    size_t stride = (size_t)gridDim.x * blockDim.x;
    for (; i < n; i += stride) c[i] = 1.0f / fmaxf(c[i], 1.0f);
}

// scatter-add raw 128-channel features of src into Asum[dst] for one relation.
// one block (128 threads) per edge; channel-coalesced atomics.
__global__ void scatter_rel_kernel(const float* __restrict__ X,
                                   const int* __restrict__ src,
                                   const int* __restrict__ dst,
                                   const int* __restrict__ et,
                                   float* __restrict__ Asum, int rel)
{
    int e = blockIdx.x;
    if (et[e] != rel) return;
    int s = src[e], d = dst[e];
    int c = threadIdx.x;   // blockDim == 128
    atomicAdd(&Asum[(size_t)d * 128 + c], X[(size_t)s * 128 + c]);
}

// out[p] = dot(z[s], fc_w[0:64]) + dot(z[d], fc_w[64:128]) + fc_b
// one wave per pair, wave32 shuffle reduction.
__global__ void decode_kernel(const float* __restrict__ z,
                              const int* __restrict__ eli,
                              const float* __restrict__ fc_w,
                              const float* __restrict__ fc_b,
                              float* __restrict__ out, int P)
{
    int wave = (int)((blockIdx.x * (size_t)blockDim.x + threadIdx.x) >> 5);
    int lane = threadIdx.x & 31;
    if (wave >= P) return;
    int s = eli[wave];
    int d = eli[P + wave];
    float sum = 0.f;
    for (int c = lane; c < 64; c += 32) {
        sum += z[(size_t)s * 64 + c] * fc_w[c];
        sum += z[(size_t)d * 64 + c] * fc_w[64 + c];
    }
    for (int off = 16; off; off >>= 1) sum += __shfl_xor(sum, off, 32);
    if (lane == 0) out[wave] = sum + fc_b[0];
}

// ---------------------------------------------------------------------------

static inline size_t align256(size_t x) { return (x + 255) & ~(size_t)255; }

extern "C" void kernel_launch(void* const* d_in, const int* in_sizes, int n_in,
                              void* d_out, int out_size, void* d_ws, size_t ws_size,
                              hipStream_t stream)
{
    (void)in_sizes; (void)n_in; (void)out_size; (void)ws_size;

    const float* x0   = (const float*)d_in[0];
    const float* x1   = (const float*)d_in[1];
    const float* x2   = (const float*)d_in[2];
    const int*   ei   = (const int*)d_in[3];     // [2, E]
    const int*   et   = (const int*)d_in[4];     // [E]
    const int*   eli  = (const int*)d_in[5];     // [2, P]
    const float* lw0  = (const float*)d_in[6];
    const float* lb0  = (const float*)d_in[7];
    const float* lw1  = (const float*)d_in[8];
    const float* lb1  = (const float*)d_in[9];
    const float* lw2  = (const float*)d_in[10];
    const float* lb2  = (const float*)d_in[11];
    const float* bas1 = (const float*)d_in[12];  // [30,128,128]
    const float* cmp1 = (const float*)d_in[13];  // [8,30]
    const float* rt1  = (const float*)d_in[14];  // [128,128]
    const float* bi1  = (const float*)d_in[15];
    const float* bas2 = (const float*)d_in[16];  // [30,128,64]
    const float* cmp2 = (const float*)d_in[17];
    const float* rt2  = (const float*)d_in[18];  // [128,64]
    const float* bi2  = (const float*)d_in[19];
    const float* fcw  = (const float*)d_in[20];  // [128,1]
    const float* fcb  = (const float*)d_in[21];
    float*       out  = (float*)d_out;

    const int* src = ei;
    const int* dst = ei + N_EDGES;

    // ---- workspace layout (~183 MB) ----
    char* base = (char*)d_ws;
    size_t off = 0;
    float* X    = (float*)(base + off); off = align256(off + (size_t)N_NODES * 128 * 4); // 51.2 MB
    float* H    = (float*)(base + off); off = align256(off + (size_t)N_NODES * 128 * 4); // 51.2 MB
    float* Z    = (float*)(base + off); off = align256(off + (size_t)N_NODES * 64  * 4); // 25.6 MB
    float* Asum = (float*)(base + off); off = align256(off + (size_t)N_NODES * 128 * 4); // 51.2 MB
    float* icnt = (float*)(base + off); off = align256(off + (size_t)N_REL * N_NODES * 4); // 3.2 MB
    float* W1   = (float*)(base + off); off = align256(off + (size_t)N_REL * 128 * 128 * 4);
    float* W2   = (float*)(base + off); off = align256(off + (size_t)N_REL * 128 * 64  * 4);

    const dim3 TB(128);
    const size_t NF128 = (size_t)N_NODES * 128;

    // ---- per-type input projections -> X [100000,128] ----
    gemm_wmma_f32<8><<<dim3((40000 + 63) / 64), TB, 0, stream>>>(x0, lw0, lb0, nullptr, X,                 40000, 256, 0, 0);
    gemm_wmma_f32<8><<<dim3((30000 + 63) / 64), TB, 0, stream>>>(x1, lw1, lb1, nullptr, X + 40000ull * 128, 30000, 128, 0, 0);
    gemm_wmma_f32<8><<<dim3((30000 + 63) / 64), TB, 0, stream>>>(x2, lw2, lb2, nullptr, X + 70000ull * 128, 30000,  64, 0, 0);

    // ---- edge counts -> inverse means (shared by both convs) ----
    fill_zero<<<dim3(2048), dim3(256), 0, stream>>>(icnt, (size_t)N_REL * N_NODES);
    count_kernel<<<dim3((N_EDGES + 255) / 256), dim3(256), 0, stream>>>(dst, et, icnt, N_EDGES);
    invert_kernel<<<dim3(2048), dim3(256), 0, stream>>>(icnt, (size_t)N_REL * N_NODES);

    // ---- relation weights from bases ----
    compute_rel_weights<<<dim3((N_REL * 128 * 128 + 255) / 256), dim3(256), 0, stream>>>(cmp1, bas1, W1, N_REL, N_BASES, 128 * 128);
    compute_rel_weights<<<dim3((N_REL * 128 * 64  + 255) / 256), dim3(256), 0, stream>>>(cmp2, bas2, W2, N_REL, N_BASES, 128 * 64);

    // ---- conv1: H = relu( X@root1 + bias1 + sum_r mean_r(X) @ W1_r ) ----
    gemm_wmma_f32<8><<<dim3((N_NODES + 63) / 64), TB, 0, stream>>>(X, rt1, bi1, nullptr, H, N_NODES, 128, 0, 0);
    for (int r = 0; r < N_REL; ++r) {
        fill_zero<<<dim3(4096), dim3(256), 0, stream>>>(Asum, NF128);
        scatter_rel_kernel<<<dim3(N_EDGES), TB, 0, stream>>>(X, src, dst, et, Asum, r);
        gemm_wmma_f32<8><<<dim3((N_NODES + 63) / 64), TB, 0, stream>>>(
            Asum, W1 + (size_t)r * 128 * 128, nullptr, icnt + (size_t)r * N_NODES,
            H, N_NODES, 128, /*accumulate=*/1, /*relu=*/(r == N_REL - 1) ? 1 : 0);
    }

    // ---- conv2: Z = H@root2 + bias2 + sum_r mean_r(H) @ W2_r ----
    gemm_wmma_f32<4><<<dim3((N_NODES + 63) / 64), TB, 0, stream>>>(H, rt2, bi2, nullptr, Z, N_NODES, 128, 0, 0);
    for (int r = 0; r < N_REL; ++r) {
        fill_zero<<<dim3(4096), dim3(256), 0, stream>>>(Asum, NF128);
        scatter_rel_kernel<<<dim3(N_EDGES), TB, 0, stream>>>(H, src, dst, et, Asum, r);
        gemm_wmma_f32<4><<<dim3((N_NODES + 63) / 64), TB, 0, stream>>>(
            Asum, W2 + (size_t)r * 128 * 64, nullptr, icnt + (size_t)r * N_NODES,
            Z, N_NODES, 128, /*accumulate=*/1, /*relu=*/0);
    }

    // ---- decode: one wave per label pair ----
    decode_kernel<<<dim3((N_PAIRS * 32 + 255) / 256), dim3(256), 0, stream>>>(Z, eli, fcw, fcb, out, N_PAIRS);
}